// EmRouting2d_3917010174025
// MI455X (gfx1250) — compile-verified
//
#include <hip/hip_runtime.h>
#include <math.h>

typedef __attribute__((ext_vector_type(2))) float v2f;
typedef __attribute__((ext_vector_type(8))) float v8f;

#define A_CAPS 32
#define B_CAPS 32
#define PSIZE  16
#define MD     4
#define KKA    288           // 3*3*32 children
#define H_IN   14
#define W_IN   14
#define OW     7
#define L_OUT  49
#define NBATCH 8
#define BL     (NBATCH*L_OUT)    // 392 (b,l) positions
#define MROWS  (BL*MD)           // 1568 GEMM rows
#define MTILES (MROWS/16)        // 98
#define NTILES ((B_CAPS*MD)/16)  // 8
#define EPS    1e-12f
#define LN_2PI 1.8378770664093453f

// v is stored internally as v[bl][k][c][q] with q = m*4 + i  (i = pose row,
// m = pose col). This makes the WMMA C-fragment store contiguous per lane
// (b128 stores) and the routing-side stream b128 loads. All reductions over
// the 16 pose dims are permutation-invariant; only the final pose_out write
// unscrambles q -> p = i*4+m.

// ---------------------------------------------------------------------------
// Kernel 1: votes via fp32 WMMA 16x16x4 (contraction dim is exactly MD=4).
// One wave per (mtile, k); A fragment (unfolded pose, zero-padded) is loaded
// once and reused across the 8 N-tiles.
// ---------------------------------------------------------------------------
__global__ __launch_bounds__(256)
void votes_wmma_kernel(const float* __restrict__ pose,
                       const float* __restrict__ Wmat,
                       float* __restrict__ vws) {
  const int lane = threadIdx.x & 31;
  const int wid  = blockIdx.x * (blockDim.x >> 5) + (threadIdx.x >> 5);
  const int mtile = wid % MTILES;
  const int k     = wid / MTILES;          // child index 0..287
  const int j0    = (lane < 16) ? 0 : 2;   // contraction dim split across halves

  // ---- A fragment: row M = mtile*16 + lane%16 -> (bl, i) ----
  const int Mrow = mtile * 16 + (lane & 15);
  const int bl = Mrow >> 2;
  const int i  = Mrow & 3;
  const int b  = bl / L_OUT;
  const int l  = bl % L_OUT;
  const int oy = l / OW, ox = l % OW;
  const int kpos = k >> 5;                 // k = kpos*32 + a
  const int a    = k & 31;
  const int ki = kpos / 3, kj = kpos % 3;
  const int h = ki + 2 * oy - 1;           // pad = 1, stride = 2
  const int w = kj + 2 * ox - 1;
  float a0 = 0.f, a1 = 0.f;
  if (h >= 0 && h < H_IN && w >= 0 && w < W_IN) {
    const size_t pb =
        (((size_t)b * (A_CAPS * PSIZE) + (size_t)(a * PSIZE + i * MD + j0)) * H_IN + h) * W_IN + w;
    a0 = pose[pb];
    a1 = pose[pb + (size_t)H_IN * W_IN];
  }
  v2f afrag; afrag.x = a0; afrag.y = a1;

  // Per-lane store rows: lanes 0-15 own M = r (r=0..7), lanes 16-31 own M = r+8.
  // With 16 rows = 4 bl positions, acc[0..3] -> blA rows i=0..3 (contiguous q),
  // acc[4..7] -> blA+1.
  const int blA = mtile * 4 + ((lane < 16) ? 0 : 2);

  #pragma unroll
  for (int nt = 0; nt < NTILES; ++nt) {
    const int Ncol = nt * 16 + (lane & 15);  // col N -> (c, m)
    const int c = Ncol >> 2;
    const int m = Ncol & 3;
    const size_t wb = (((size_t)k * B_CAPS + c) * PSIZE) + (size_t)(j0 * MD + m);
    v2f bfrag; bfrag.x = Wmat[wb]; bfrag.y = Wmat[wb + MD];

    v8f acc = {};
    acc = __builtin_amdgcn_wmma_f32_16x16x4_f32(false, afrag, false, bfrag,
                                                (short)0, acc, false, false);

    const size_t baseA = (((size_t)blA * KKA + k) * B_CAPS + c) * PSIZE + (size_t)(m * MD);
    const size_t baseB = (((size_t)(blA + 1) * KKA + k) * B_CAPS + c) * PSIZE + (size_t)(m * MD);
    float4 fa; fa.x = acc[0]; fa.y = acc[1]; fa.z = acc[2]; fa.w = acc[3];
    float4 fb; fb.x = acc[4]; fb.y = acc[5]; fb.z = acc[6]; fb.w = acc[7];
    *reinterpret_cast<float4*>(vws + baseA) = fa;   // global_store_b128
    *reinterpret_cast<float4*>(vws + baseB) = fb;   // global_store_b128
  }
}

// ---------------------------------------------------------------------------
// Kernel 2: EM routing. One 256-thread block per (b,l) position; r[288][32]
// and all statistics LDS-resident; cross-child reductions via ds_add_f32
// atomics; v streamed from global as b128 loads.
// Thread mapping: c = tid & 31 (parent), g = tid >> 5 (8 groups of 36 k's).
// ---------------------------------------------------------------------------
__global__ __launch_bounds__(256)
void em_routing_kernel(const float* __restrict__ a_in,
                       const float* __restrict__ vws,
                       const float* __restrict__ beta_u,
                       const float* __restrict__ beta_a,
                       float* __restrict__ out) {
  __shared__ float sh_r[KKA * B_CAPS];     // routing weights / ln_ap scratch
  __shared__ float sh_au[KKA];             // unfolded activations
  __shared__ float sh_mu[B_CAPS * PSIZE];  // indexed by q
  __shared__ float sh_sig[B_CAPS * PSIZE]; // indexed by q
  __shared__ float sh_rsum[B_CAPS];
  __shared__ float sh_aout[B_CAPS];

  const int tid = threadIdx.x;
  const int bl  = blockIdx.x;
  const int b = bl / L_OUT;
  const int l = bl % L_OUT;
  const int oy = l / OW, ox = l % OW;

  // unfold a_in into LDS; init r = 1
  for (int k = tid; k < KKA; k += 256) {
    const int kpos = k >> 5;
    const int a = k & 31;
    const int ki = kpos / 3, kj = kpos % 3;
    const int h = ki + 2 * oy - 1;
    const int w = kj + 2 * ox - 1;
    float av = 0.f;
    if (h >= 0 && h < H_IN && w >= 0 && w < W_IN)
      av = a_in[(((size_t)b * A_CAPS + a) * H_IN + h) * W_IN + w];
    sh_au[k] = av;
  }
  for (int x = tid; x < KKA * B_CAPS; x += 256) sh_r[x] = 1.0f;
  __syncthreads();

  const int c  = tid & 31;
  const int g  = tid >> 5;
  const int k0 = g * (KKA / 8);           // 36 children per thread
  const float lam_tab[3] = {5.0e-4f, 9.75e-4f, 1.42625e-3f};  // 0.01*(1-0.95^(i+1))

  for (int it = 0; it < 3; ++it) {
    for (int x = tid; x < B_CAPS * PSIZE; x += 256) { sh_mu[x] = 0.f; sh_sig[x] = 0.f; }
    if (tid < B_CAPS) sh_rsum[tid] = 0.f;
    __syncthreads();

    // r_sum[c] = sum_k r*au
    float part = 0.f;
    for (int kk = 0; kk < KKA / 8; ++kk) {
      const int k = k0 + kk;
      part += sh_r[k * B_CAPS + c] * sh_au[k];
    }
    atomicAdd(&sh_rsum[c], part);
    __syncthreads();
    const float rs = sh_rsum[c] + EPS;

    // mu[c,q] = sum_k coeff * v   (b128 streamed)
    float pmu[PSIZE];
    #pragma unroll
    for (int q = 0; q < PSIZE; ++q) pmu[q] = 0.f;
    for (int kk = 0; kk < KKA / 8; ++kk) {
      const int k = k0 + kk;
      const float coeff = sh_r[k * B_CAPS + c] * sh_au[k] / rs;
      const float4* vp =
          reinterpret_cast<const float4*>(vws + (((size_t)bl * KKA + k) * B_CAPS + c) * PSIZE);
      #pragma unroll
      for (int j = 0; j < 4; ++j) {
        const float4 v4 = vp[j];
        pmu[j * 4 + 0] += coeff * v4.x;
        pmu[j * 4 + 1] += coeff * v4.y;
        pmu[j * 4 + 2] += coeff * v4.z;
        pmu[j * 4 + 3] += coeff * v4.w;
      }
    }
    #pragma unroll
    for (int q = 0; q < PSIZE; ++q) atomicAdd(&sh_mu[c * PSIZE + q], pmu[q]);
    __syncthreads();

    // sigma[c,q] = sum_k coeff * (v - mu)^2
    float psg[PSIZE];
    #pragma unroll
    for (int q = 0; q < PSIZE; ++q) psg[q] = 0.f;
    for (int kk = 0; kk < KKA / 8; ++kk) {
      const int k = k0 + kk;
      const float coeff = sh_r[k * B_CAPS + c] * sh_au[k] / rs;
      const float4* vp =
          reinterpret_cast<const float4*>(vws + (((size_t)bl * KKA + k) * B_CAPS + c) * PSIZE);
      #pragma unroll
      for (int j = 0; j < 4; ++j) {
        const float4 v4 = vp[j];
        const float d0 = v4.x - sh_mu[c * PSIZE + j * 4 + 0];
        const float d1 = v4.y - sh_mu[c * PSIZE + j * 4 + 1];
        const float d2 = v4.z - sh_mu[c * PSIZE + j * 4 + 2];
        const float d3 = v4.w - sh_mu[c * PSIZE + j * 4 + 3];
        psg[j * 4 + 0] += coeff * d0 * d0;
        psg[j * 4 + 1] += coeff * d1 * d1;
        psg[j * 4 + 2] += coeff * d2 * d2;
        psg[j * 4 + 3] += coeff * d3 * d3;
      }
    }
    #pragma unroll
    for (int q = 0; q < PSIZE; ++q) atomicAdd(&sh_sig[c * PSIZE + q], psg[q]);
    __syncthreads();

    // cost & output activation (sum over pose dims is order-invariant)
    if (tid < B_CAPS) {
      float cost = 0.f;
      const float bu  = beta_u[tid];
      const float rsl = sh_rsum[tid];        // reference uses raw r_sum here
      for (int q = 0; q < PSIZE; ++q) {
        const float sg = sh_sig[tid * PSIZE + q] + EPS;
        cost += (bu + 0.5f * __logf(sg)) * rsl;
      }
      const float z = lam_tab[it] * (beta_a[tid] - cost);
      sh_aout[tid] = 1.f / (1.f + __expf(-z));
    }
    __syncthreads();

    if (it < 2) {
      // ln_ap = ln_p + log(a_out) -> overwrite sh_r, then softmax over parents
      const float ln_a = __logf(sh_aout[c]);
      for (int kk = 0; kk < KKA / 8; ++kk) {
        const int k = k0 + kk;
        const float4* vp =
            reinterpret_cast<const float4*>(vws + (((size_t)bl * KKA + k) * B_CAPS + c) * PSIZE);
        float s1 = 0.f, s2 = 0.f;
        #pragma unroll
        for (int j = 0; j < 4; ++j) {
          const float4 v4 = vp[j];
          const float vv[4] = {v4.x, v4.y, v4.z, v4.w};
          #pragma unroll
          for (int jj = 0; jj < 4; ++jj) {
            const int q = j * 4 + jj;
            const float sg = sh_sig[c * PSIZE + q] + EPS;
            const float d  = vv[jj] - sh_mu[c * PSIZE + q];
            s1 += __logf(sg * LN_2PI);
            s2 += d * d / (2.f * sg);
          }
        }
        sh_r[k * B_CAPS + c] = -0.5f * s1 - s2 + ln_a;
      }
      __syncthreads();
      for (int k = tid; k < KKA; k += 256) {
        float mx = -3.4e38f;
        for (int cc = 0; cc < B_CAPS; ++cc) mx = fmaxf(mx, sh_r[k * B_CAPS + cc]);
        float s = 0.f;
        for (int cc = 0; cc < B_CAPS; ++cc) s += __expf(sh_r[k * B_CAPS + cc] - mx);
        const float inv = 1.f / s;
        for (int cc = 0; cc < B_CAPS; ++cc)
          sh_r[k * B_CAPS + cc] = __expf(sh_r[k * B_CAPS + cc] - mx) * inv;
      }
      __syncthreads();
    }
  }

  // outputs: a_o (b,B,7,7) then pose_out (b,512,7,7), concatenated flat.
  // pose channel uses p = i*4+m; internal q = m*4+i -> p = (q&3)*4 + (q>>2).
  if (tid < B_CAPS)
    out[((size_t)b * B_CAPS + tid) * L_OUT + l] = sh_aout[tid];
  for (int x = tid; x < B_CAPS * PSIZE; x += 256) {
    const int cc = x >> 4;
    const int q  = x & 15;
    const int p  = (q & 3) * 4 + (q >> 2);
    out[(size_t)NBATCH * B_CAPS * L_OUT +
        ((size_t)b * (B_CAPS * PSIZE) + (size_t)(cc * PSIZE + p)) * L_OUT + l] = sh_mu[x];
  }
}

// ---------------------------------------------------------------------------
extern "C" void kernel_launch(void* const* d_in, const int* in_sizes, int n_in,
                              void* d_out, int out_size, void* d_ws, size_t ws_size,
                              hipStream_t stream) {
  const float* a_in   = (const float*)d_in[0];   // (8,32,14,14)
  const float* pose   = (const float*)d_in[1];   // (8,512,14,14)
  const float* Wmat   = (const float*)d_in[2];   // (288,32,4,4)
  const float* beta_u = (const float*)d_in[3];   // (32,)
  const float* beta_a = (const float*)d_in[4];   // (32,)
  float* out = (float*)d_out;
  float* vws = (float*)d_ws;                     // votes: 392*288*32*16 f32 = 231 MB

  // 98 M-tiles * 288 children = 28224 waves, 8 waves/block -> 3528 blocks
  votes_wmma_kernel<<<(MTILES * KKA) / 8, 256, 0, stream>>>(pose, Wmat, vws);
  // one block per (b,l) position
  em_routing_kernel<<<BL, 256, 0, stream>>>(a_in, vws, beta_u, beta_a, out);
}